// Gridding_39891656245673
// MI455X (gfx1250) — compile-verified
//
#include <hip/hip_runtime.h>
#include <cstdint>

#define GRID_D 64           // SCALES = (64,64,64)
#define HALF_D 32
#define CELLS  (GRID_D * GRID_D * GRID_D)

// ---- gfx1250 async-to-LDS feature detection (guarded: falls back cleanly) ----
#if defined(__has_builtin)
#  if __has_builtin(__builtin_amdgcn_global_load_async_to_lds_b32)
#    define HAVE_ASYNC_COPY 1
#  endif
#  if __has_builtin(__builtin_amdgcn_s_wait_asynccnt)
#    define HAVE_WAIT_ASYNC 1
#  endif
#endif
#ifndef HAVE_ASYNC_COPY
#  define HAVE_ASYNC_COPY 0
#endif
#ifndef HAVE_WAIT_ASYNC
#  define HAVE_WAIT_ASYNC 0
#endif

// The builtin takes typed int pointers: AS(1) global source, AS(3) LDS dest.
typedef __attribute__((address_space(1))) int* as1_i32;   // global (__device__ int*)
typedef __attribute__((address_space(3))) int* as3_i32;   // LDS    (__shared__ int*)

// Zero-initialize the output grid with wide 128-bit stores.
// Launched with exactly ceil(n4)/256 blocks: one float4 per thread, no loop.
__global__ __launch_bounds__(256) void zero_f32x4(float4* __restrict__ out, int n4) {
  const int i = blockIdx.x * 256 + threadIdx.x;
  if (i < n4) out[i] = make_float4(0.0f, 0.0f, 0.0f, 0.0f);  // global_store_b128
}

// Scalar tail for out_size % 4 != 0 (not hit for 64^3 grids; kept for generality).
__global__ void zero_f32_tail(float* __restrict__ out, int start, int n) {
  const int i = start + blockIdx.x * 256 + threadIdx.x;
  if (i < n) out[i] = 0.0f;
}

// One thread per point; 256-point tile staged to LDS per block.
__global__ __launch_bounds__(256) void gridding_scatter(
    const float* __restrict__ pts,   // (B*N, 3) flattened
    float* __restrict__ grid,        // (B, 64*64*64) flattened
    int npts, int N) {
  __shared__ float tile[256 * 3];    // 3 KiB point tile

  const int tid   = threadIdx.x;
  const int base  = blockIdx.x << 8;   // first point of this block
  const int dbase = base * 3;          // first dword of this block's tile
  const int total_dw = npts * 3;

#if HAVE_ASYNC_COPY
  // gfx1250 path: GLOBAL_LOAD_ASYNC_TO_LDS_B32, tracked by ASYNCcnt.
  #pragma unroll
  for (int r = 0; r < 3; ++r) {
    const int off = (r << 8) + tid;
    if (dbase + off < total_dw) {
      __builtin_amdgcn_global_load_async_to_lds_b32(
          (as1_i32)(uintptr_t)(pts + dbase + off),
          (as3_i32)(unsigned)(uintptr_t)&tile[off],
          /*offset=*/0, /*cpol=*/0);
    }
  }
#  if HAVE_WAIT_ASYNC
  __builtin_amdgcn_s_wait_asynccnt(0);
#  else
  asm volatile("s_wait_asynccnt 0" ::: "memory");
#  endif
#else
  // Fallback: coalesced direct loads through VGPRs into LDS.
  #pragma unroll
  for (int r = 0; r < 3; ++r) {
    const int off = (r << 8) + tid;
    if (dbase + off < total_dw) tile[off] = pts[dbase + off];
  }
#endif
  __syncthreads();

  const int p = base + tid;
  if (p >= npts) return;

  // Batch index: block-uniform (SALU) when N is a multiple of the 256-pt tile.
  int b;
  if ((N & 255) == 0) b = base / N;   // uniform -> one scalar division per block
  else                b = p / N;      // general fallback
  const int bOff = b * CELLS;

  // Scale by sx/2 == 32 for all three dims.
  const float px = tile[tid * 3 + 0] * 32.0f;
  const float py = tile[tid * 3 + 1] * 32.0f;
  const float pz = tile[tid * 3 + 2] * 32.0f;

  // nz_mask: sum(|p|) != 0
  if (fabsf(px) + fabsf(py) + fabsf(pz) == 0.0f) return;

  const float lx = floorf(px), ly = floorf(py), lz = floorf(pz);
  const float fx = px - lx,    fy = py - ly,    fz = pz - lz;
  const int ix = (int)lx + HALF_D;
  const int iy = (int)ly + HALF_D;
  const int iz = (int)lz + HALF_D;

  const float wx[2] = {1.0f - fx, fx};
  const float wy[2] = {1.0f - fy, fy};
  const float wz[2] = {1.0f - fz, fz};
  const int   X[2]  = {ix, ix + 1};
  const int   Y[2]  = {iy, iy + 1};
  const int   Z[2]  = {iz, iz + 1};

  // 8 trilinear scatter-adds; skip out-of-bounds corners entirely
  // (equivalent to the reference's zeroed-weight + clipped-index add).
  #pragma unroll
  for (int i = 0; i < 2; ++i) {
    if ((unsigned)X[i] < (unsigned)GRID_D) {
      const int xo = bOff + X[i] * (GRID_D * GRID_D);
      #pragma unroll
      for (int j = 0; j < 2; ++j) {
        if ((unsigned)Y[j] < (unsigned)GRID_D) {
          const int yo = xo + Y[j] * GRID_D;
          const float wxy = wx[i] * wy[j];
          #pragma unroll
          for (int k = 0; k < 2; ++k) {
            if ((unsigned)Z[k] < (unsigned)GRID_D) {
              atomicAdd(&grid[yo + Z[k]], wxy * wz[k]);  // global_atomic_add_f32
            }
          }
        }
      }
    }
  }
}

extern "C" void kernel_launch(void* const* d_in, const int* in_sizes, int n_in,
                              void* d_out, int out_size, void* d_ws, size_t ws_size,
                              hipStream_t stream) {
  const float* pts = (const float*)d_in[0];
  float* out = (float*)d_out;

  const int npts = in_sizes[0] / 3;           // B*N = 4,194,304
  const int B    = out_size / CELLS;          // 64
  const int N    = (B > 0) ? (npts / B) : 1;  // 65536

  // 1) zero the grid: wide b128 stores (64 MiB -> ~3 us at HBM bw, L2-resident after)
  const int n4 = out_size >> 2;
  if (n4 > 0) {
    const int zblocks = (n4 + 255) / 256;
    zero_f32x4<<<zblocks, 256, 0, stream>>>((float4*)out, n4);
  }
  const int tail = out_size & 3;
  if (tail) {
    zero_f32_tail<<<1, 256, 0, stream>>>(out, n4 << 2, out_size);
  }

  // 2) scatter: one thread per point
  const int sblocks = (npts + 255) / 256;
  gridding_scatter<<<sblocks, 256, 0, stream>>>(pts, out, npts, N);
}